// MultiHeadAttention_33131377721819
// MI455X (gfx1250) — compile-verified
//
#include <hip/hip_runtime.h>

typedef __attribute__((ext_vector_type(2))) float v2f;
typedef __attribute__((ext_vector_type(4))) float v4f;
typedef __attribute__((ext_vector_type(8))) float v8f;
typedef __attribute__((ext_vector_type(4))) unsigned int u32x4;
typedef __attribute__((ext_vector_type(8))) int i32x8;
typedef __attribute__((ext_vector_type(4))) int i32x4;

// Problem constants
#define EMB   768
#define NSEQ  1024
#define HEADS 12
#define DHEAD 64
#define BATCH 8
#define MROWS (BATCH * NSEQ)          // 8192
#define HSTRIDE (NSEQ * DHEAD)        // 65536  (flat per-head stride from the faithful reshape)
#define BSTRIDE (NSEQ * EMB)          // 786432 (per-batch stride)

static __device__ __forceinline__ v8f wmma_f32(v2f a, v2f b, v8f c) {
    return __builtin_amdgcn_wmma_f32_16x16x4_f32(false, a, false, b, (short)0, c, false, false);
}

// ---------------------------------------------------------------------------
// TDM: issue a 2D f32 tile load (tile_h rows x tile_w cols, row stride
// `stride` elements) from global into contiguous LDS at byte offset lds_addr.
// D# bitfields per CDNA5 ISA.
// This toolchain exposes the 6-arg builtin: (g0, g1, g2, g3, g4, cpol).
// ---------------------------------------------------------------------------
static __device__ __forceinline__ void tdm_load_tile_f32(
    unsigned lds_addr, const float* gaddr,
    unsigned tile_w, unsigned tile_h,
    unsigned tensor_w, unsigned tensor_h, unsigned stride)
{
    unsigned long long ga = (unsigned long long)(uintptr_t)gaddr;
    u32x4 g0;
    g0[0] = 1u;                                   // count=1, is_restore=0, gather=0
    g0[1] = lds_addr;                             // lds_addr[31:0] (bytes)
    g0[2] = (unsigned)(ga & 0xffffffffu);         // global_addr[31:0]
    g0[3] = (unsigned)((ga >> 32) & 0x01ffffffu)  // global_addr[56:32]
          | (2u << 30);                           // type = 2 ("image")
    i32x8 g1;
    g1[0] = (int)(2u << 16);                      // data_size=2 (4 bytes); mask/flags 0
    g1[1] = (int)((tensor_w & 0xffffu) << 16);    // [31:16] = tensor_dim0[15:0]
    g1[2] = (int)((tensor_w >> 16) | ((tensor_h & 0xffffu) << 16)); // dim0 hi | dim1 lo
    g1[3] = (int)((tensor_h >> 16) | (tile_w << 16));               // dim1 hi | tile_dim0
    g1[4] = (int)(tile_h & 0xffffu);              // tile_dim1 (tile_dim2 = 0)
    g1[5] = (int)stride;                          // tensor_dim0_stride[31:0]
    g1[6] = 0;                                    // stride0[47:32] | stride1[15:0]
    g1[7] = 0;                                    // stride1[47:16]
    const i32x4 z4 = {0, 0, 0, 0};                // groups 2/3 unused (<=2D)
    const i32x8 z8 = {0, 0, 0, 0, 0, 0, 0, 0};    // extra group (unused)
    __builtin_amdgcn_tensor_load_to_lds(g0, g1, z4, z4, z8, 0);
}

// ---------------------------------------------------------------------------
// GEMM: Y[M,768] = X[M,768] @ W[768,768], f32 WMMA 16x16x4.
// Block = 128 threads (4 waves), 64x64 C tile. TDM double-buffered staging:
// wave 0 issues chunk c+1's two tile DMAs before waiting on chunk c
// (per-wave TDM completion is in-order), overlapping DMA with WMMA.
// ---------------------------------------------------------------------------
__global__ __launch_bounds__(128)
void gemm_f32_wmma(const float* __restrict__ X, const float* __restrict__ W,
                   float* __restrict__ Y) {
    __shared__ float Sm[4 * 1024];  // As[2][64*16] @ 0, Bs[2][16*64] @ 2048 (floats)

    const int tid  = threadIdx.x;
    const int w    = tid >> 5;
    const int lane = tid & 31;
    const int l16  = lane & 15;
    const int hi   = lane >> 4;          // 0: lanes 0-15 (K+0/1), 1: lanes 16-31 (K+2/3)
    const int blk  = blockIdx.x;
    const int m0   = (blk & 127) * 64;   // M/64 = 128
    const int n0   = (blk >> 7) * 64;    // 768/64 = 12

    const unsigned lds_base = (unsigned)(uintptr_t)&Sm[0];

    v8f acc[4];
    const v8f zero = {0.f, 0.f, 0.f, 0.f, 0.f, 0.f, 0.f, 0.f};
#pragma unroll
    for (int j = 0; j < 4; ++j) acc[j] = zero;

    const int NCHUNK = EMB / 16;         // 48
    if (w == 0) {                        // prime the pipeline: chunk 0 -> buffer 0
        tdm_load_tile_f32(lds_base, X + (size_t)m0 * EMB, 16, 64, EMB, MROWS, EMB);
        tdm_load_tile_f32(lds_base + 2048u * 4u, W + n0, 64, 16, EMB, EMB, EMB);
    }
    for (int c = 0; c < NCHUNK; ++c) {
        const int buf  = c & 1;
        if (w == 0) {
            if (c + 1 < NCHUNK) {        // issue next chunk into the other buffer
                const int nb = (c + 1) & 1;
                const int k1 = (c + 1) * 16;
                tdm_load_tile_f32(lds_base + (unsigned)nb * 4096u,
                                  X + (size_t)m0 * EMB + k1, 16, 64, EMB, MROWS, EMB);
                tdm_load_tile_f32(lds_base + 8192u + (unsigned)nb * 4096u,
                                  W + (size_t)k1 * EMB + n0, 64, 16, EMB, EMB, EMB);
                __builtin_amdgcn_s_wait_tensorcnt(2);  // chunk c complete (in-order)
            } else {
                __builtin_amdgcn_s_wait_tensorcnt(0);
            }
        }
        __syncthreads();                 // chunk c tiles visible to all waves
        const float* As = &Sm[buf * 1024];
        const float* Bs = &Sm[2048 + buf * 1024];
#pragma unroll
        for (int kk = 0; kk < 4; ++kk) {
            v2f a = *(const v2f*)&As[(w * 16 + l16) * 16 + kk * 4 + hi * 2];
#pragma unroll
            for (int j = 0; j < 4; ++j) {
                v2f b;
                b.x = Bs[(kk * 4 + hi * 2 + 0) * 64 + j * 16 + l16];
                b.y = Bs[(kk * 4 + hi * 2 + 1) * 64 + j * 16 + l16];
                acc[j] = wmma_f32(a, b, acc[j]);
            }
        }
        __syncthreads();                 // all reads of buf done before TDM reuses it
    }
    // C layout: VGPR v -> row v (lanes 0-15) / row v+8 (lanes 16-31), col = l16.
#pragma unroll
    for (int j = 0; j < 4; ++j)
#pragma unroll
        for (int v = 0; v < 8; ++v) {
            const int row = m0 + w * 16 + v + hi * 8;
            const int col = n0 + j * 16 + l16;
            Y[(size_t)row * EMB + col] = acc[j][v];
        }
}

// ---------------------------------------------------------------------------
// Attention: one block (8 waves, 256 thr) per (b, h, 16-query-row block).
// S row-block (16x1024 f32 = 64KB) kept entirely in LDS.
// ---------------------------------------------------------------------------
__global__ __launch_bounds__(256)
void attn_kernel(const float* __restrict__ Q, const float* __restrict__ K,
                 const float* __restrict__ V, const float* __restrict__ Bm,
                 float* __restrict__ Hc) {
    __shared__ float S[16 * NSEQ];       // 64 KB

    const int tid  = threadIdx.x;
    const int w    = tid >> 5;
    const int lane = tid & 31;
    const int l16  = lane & 15;
    const int hi   = lane >> 4;

    int blk = blockIdx.x;
    const int rb = blk & 63;  blk >>= 6; // N/16 = 64 row blocks
    const int h  = blk % HEADS;
    const int b  = blk / HEADS;
    const int r0 = rb * 16;
    const float scale = 0.125f;          // 1/sqrt(64)

    const float* Qh = Q + (size_t)b * BSTRIDE + (size_t)h * HSTRIDE; // [1024,64]
    const float* Kh = K + (size_t)b * BSTRIDE + (size_t)h * HSTRIDE;
    const float* Vh = V + (size_t)b * BSTRIDE + (size_t)h * HSTRIDE;
    const float* Bh = Bm + (size_t)h * NSEQ * NSEQ;                  // [1024,1024]

    // ---- Phase 1: S = Bm + scale * Q @ K^T.  Wave w -> key cols [128w, 128w+128).
    const int c0 = w * 128;
    v8f acc[8];
#pragma unroll
    for (int j = 0; j < 8; ++j) {
        v8f t;
#pragma unroll
        for (int v = 0; v < 8; ++v)
            t[v] = Bh[(size_t)(r0 + v + hi * 8) * NSEQ + c0 + j * 16 + l16];
        acc[j] = t;                      // bias preloaded as C accumulator
    }
#pragma unroll 4
    for (int kk = 0; kk < 16; ++kk) {    // K = 64, 4 per WMMA
        v2f a = *(const v2f*)&Qh[(size_t)(r0 + l16) * DHEAD + kk * 4 + hi * 2];
        a.x *= scale; a.y *= scale;      // fold softmax scale into A
#pragma unroll
        for (int j = 0; j < 8; ++j) {
            // B = K^T: B[k][col] = Kh[col*64 + k] -> contiguous pair along k
            v2f bf = *(const v2f*)&Kh[(size_t)(c0 + j * 16 + l16) * DHEAD + kk * 4 + hi * 2];
            acc[j] = wmma_f32(a, bf, acc[j]);
        }
    }
    // ---- Phase 2: spill S tiles to LDS.
#pragma unroll
    for (int j = 0; j < 8; ++j)
#pragma unroll
        for (int v = 0; v < 8; ++v)
            S[(v + hi * 8) * NSEQ + c0 + j * 16 + l16] = acc[j][v];
    __syncthreads();

    // ---- Phase 3: row softmax. Wave w owns rows 2w, 2w+1; lane-parallel cols.
    for (int rr = 0; rr < 2; ++rr) {
        float* row = &S[(w * 2 + rr) * NSEQ];
        float m = -3.402823e38f;
        for (int c = lane; c < NSEQ; c += 32) m = fmaxf(m, row[c]);
#pragma unroll
        for (int off = 16; off > 0; off >>= 1) m = fmaxf(m, __shfl_xor(m, off, 32));
        float s = 0.f;
        for (int c = lane; c < NSEQ; c += 32) {
            float e = __expf(row[c] - m);
            row[c] = e;
            s += e;
        }
#pragma unroll
        for (int off = 16; off > 0; off >>= 1) s += __shfl_xor(s, off, 32);
        const float inv = 1.0f / s;
        for (int c = lane; c < NSEQ; c += 32) row[c] *= inv;
    }
    __syncthreads();

    // ---- Phase 4: O = P @ V, split-K across wave pairs.
    const int t  = w & 3;                // head-dim col tile (16 cols)
    const int kh = w >> 2;               // K half: [0,512) or [512,1024)
    v8f o = {0.f, 0.f, 0.f, 0.f, 0.f, 0.f, 0.f, 0.f};
#pragma unroll 4
    for (int kk = 0; kk < 128; ++kk) {
        const int k = kh * 512 + kk * 4;
        v2f a = *(const v2f*)&S[l16 * NSEQ + k + hi * 2];  // P fragment from LDS
        v2f bf;
        bf.x = Vh[(size_t)(k + hi * 2 + 0) * DHEAD + t * 16 + l16];
        bf.y = Vh[(size_t)(k + hi * 2 + 1) * DHEAD + t * 16 + l16];
        o = wmma_f32(a, bf, o);
    }
    __syncthreads();                     // all P reads done -> safe to reuse LDS
    if (w >= 4) {
#pragma unroll
        for (int v = 0; v < 8; ++v) S[(w - 4) * 256 + v * 32 + lane] = o[v];
    }
    __syncthreads();
    if (w < 4) {
#pragma unroll
        for (int v = 0; v < 8; ++v) {
            const float val = o[v] + S[w * 256 + v * 32 + lane];
            const int row = r0 + v + hi * 8;
            const int col = h * DHEAD + t * 16 + l16;  // transpose(0,2,1,3) merge
            Hc[(size_t)b * BSTRIDE + (size_t)row * EMB + col] = val;
        }
    }
}

// ---------------------------------------------------------------------------
extern "C" void kernel_launch(void* const* d_in, const int* in_sizes, int n_in,
                              void* d_out, int out_size, void* d_ws, size_t ws_size,
                              hipStream_t stream) {
    (void)in_sizes; (void)n_in; (void)out_size; (void)ws_size;
    const float* emb = (const float*)d_in[0];
    const float* Bm  = (const float*)d_in[1];
    const float* WQ  = (const float*)d_in[2];
    const float* WK  = (const float*)d_in[3];
    const float* WV  = (const float*)d_in[4];
    const float* WO  = (const float*)d_in[5];
    float* out = (float*)d_out;

    const size_t SZ = (size_t)MROWS * EMB;   // 6.29M floats each
    float* Qb = (float*)d_ws;
    float* Kb = Qb + SZ;
    float* Vb = Kb + SZ;
    float* Hc = Vb + SZ;

    const int gemm_blocks = (MROWS / 64) * (EMB / 64);   // 1536
    gemm_f32_wmma<<<gemm_blocks, 128, 0, stream>>>(emb, WQ, Qb);
    gemm_f32_wmma<<<gemm_blocks, 128, 0, stream>>>(emb, WK, Kb);
    gemm_f32_wmma<<<gemm_blocks, 128, 0, stream>>>(emb, WV, Vb);

    const int attn_blocks = BATCH * HEADS * (NSEQ / 16); // 6144
    attn_kernel<<<attn_blocks, 256, 0, stream>>>(Qb, Kb, Vb, Bm, Hc);

    gemm_f32_wmma<<<gemm_blocks, 128, 0, stream>>>(Hc, WO, out);
}